// GNNModel_67516885893240
// MI455X (gfx1250) — compile-verified
//
#include <hip/hip_runtime.h>

typedef float v2f __attribute__((ext_vector_type(2)));
typedef float v8f __attribute__((ext_vector_type(8)));

// ---------------------------------------------------------------- degree ----
__global__ void k_fill_deg(float* __restrict__ deg, int n) {
    int i = blockIdx.x * blockDim.x + threadIdx.x;
    if (i < n) deg[i] = 1.0f;  // self-loop contributes 1
}

__global__ void k_count_deg(const int* __restrict__ col, float* __restrict__ deg,
                            int n_edges) {
    int e = blockIdx.x * blockDim.x + threadIdx.x;
    if (e < n_edges) atomicAdd(&deg[col[e]], 1.0f);
}

__global__ void k_dinv(float* __restrict__ deg, int n) {
    int i = blockIdx.x * blockDim.x + threadIdx.x;
    if (i < n) deg[i] = rsqrtf(fmaxf(deg[i], 1.0f));
}

// ------------------------------------------------- fp32 WMMA GEMM, N = 64 ----
// C[M,64] = A[M,K] @ B[K,64]; one wave computes a 16x64 tile with four
// 16x16 f32 accumulators via V_WMMA_F32_16X16X4_F32. B staged in LDS.
__global__ void k_gemm_wmma_n64(const float* __restrict__ A,
                                const float* __restrict__ B,
                                float* __restrict__ C, int M, int K) {
    extern __shared__ float sB[];  // K * 64 floats
    const int tid = threadIdx.x;

    // Stage B (K x 64) into LDS with float4 loads.
    const int nv4 = K * 16;
    const float4* Bv = (const float4*)B;
    float4* sBv = (float4*)sB;
    for (int i = tid; i < nv4; i += blockDim.x) sBv[i] = Bv[i];
    __syncthreads();

    const int wave = tid >> 5;
    const int lane = tid & 31;
    const int tile = blockIdx.x * (blockDim.x >> 5) + wave;
    if (tile * 16 >= M) return;  // wave-uniform: EXEC stays all-ones

    const int row0 = tile * 16;
    const int m    = lane & 15;       // A row within tile / B column within 16
    const int kh   = (lane >> 4) << 1;  // K sub-offset: 0 (lanes 0-15) or 2

    int arow_idx = row0 + m;
    if (arow_idx >= M) arow_idx = M - 1;  // clamp (keeps EXEC full)
    const float* arow = A + (size_t)arow_idx * K + kh;

    v8f c0 = {}, c1 = {}, c2 = {}, c3 = {};
    for (int k0 = 0; k0 < K; k0 += 4) {
        v2f a;
        a.x = arow[k0];
        a.y = arow[k0 + 1];
        const float* brow0 = sB + (k0 + kh) * 64 + m;
        const float* brow1 = sB + (k0 + kh + 1) * 64 + m;
        v2f b;
        b.x = brow0[0];  b.y = brow1[0];
        c0 = __builtin_amdgcn_wmma_f32_16x16x4_f32(false, a, false, b, (short)0, c0, false, false);
        b.x = brow0[16]; b.y = brow1[16];
        c1 = __builtin_amdgcn_wmma_f32_16x16x4_f32(false, a, false, b, (short)0, c1, false, false);
        b.x = brow0[32]; b.y = brow1[32];
        c2 = __builtin_amdgcn_wmma_f32_16x16x4_f32(false, a, false, b, (short)0, c2, false, false);
        b.x = brow0[48]; b.y = brow1[48];
        c3 = __builtin_amdgcn_wmma_f32_16x16x4_f32(false, a, false, b, (short)0, c3, false, false);
    }

    // C/D layout: VGPR r, lanes 0-15 -> row r, lanes 16-31 -> row r+8; col = lane&15.
    const int nlo  = lane & 15;
    const int rhal = (lane >> 4) * 8;
#pragma unroll
    for (int r = 0; r < 8; ++r) {
        int orow = row0 + r + rhal;
        if (orow < M) {
            float* o = C + (size_t)orow * 64 + nlo;
            o[0]  = c0[r];
            o[16] = c1[r];
            o[32] = c2[r];
            o[48] = c3[r];
        }
    }
}

// ---------------------------------------------------------- aggregation ----
// agg[i,f] = h[i,f] * dinv[i]^2   (self-loop term, also initializes agg)
__global__ void k_selfloop(const float* __restrict__ h, const float* __restrict__ dinv,
                           float* __restrict__ agg, int n) {
    int idx = blockIdx.x * blockDim.x + threadIdx.x;
    if (idx < n * 64) {
        float d = dinv[idx >> 6];
        agg[idx] = h[idx] * d * d;
    }
}

// agg[col,f] += h[row,f] * dinv[row]*dinv[col]; thread = (edge, feature);
// 64 feats/edge -> edge index is wave-uniform, h/agg accesses coalesced.
__global__ void k_scatter(const int* __restrict__ row, const int* __restrict__ col,
                          const float* __restrict__ dinv, const float* __restrict__ h,
                          float* __restrict__ agg, int n_edges) {
    long long idx = (long long)blockIdx.x * blockDim.x + threadIdx.x;
    if (idx >= (long long)n_edges * 64) return;
    int e = (int)(idx >> 6);
    int f = (int)(idx & 63);
    int r = row[e];
    int c = col[e];
    float v = h[(size_t)r * 64 + f] * dinv[r] * dinv[c];
    atomicAdd(&agg[(size_t)c * 64 + f], v);
}

__global__ void k_bias_relu(float* __restrict__ h, const float* __restrict__ b, int n) {
    int idx = blockIdx.x * blockDim.x + threadIdx.x;
    if (idx < n * 64) h[idx] = fmaxf(h[idx] + b[idx & 63], 0.0f);
}

// --------------------------------------------------------- final 64->4 FC ----
__global__ void k_fc(const float* __restrict__ h, const float* __restrict__ W,
                     const float* __restrict__ b, float* __restrict__ out, int n) {
    int node = blockIdx.x * blockDim.x + threadIdx.x;
    if (node >= n) return;
    float a0 = b[0], a1 = b[1], a2 = b[2], a3 = b[3];
    const float* hr = h + (size_t)node * 64;
#pragma unroll
    for (int j = 0; j < 64; ++j) {
        float x = hr[j];
        a0 += x * W[j * 4 + 0];
        a1 += x * W[j * 4 + 1];
        a2 += x * W[j * 4 + 2];
        a3 += x * W[j * 4 + 3];
    }
    float* o = out + (size_t)node * 4;
    o[0] = a0; o[1] = a1; o[2] = a2; o[3] = a3;
}

// -------------------------------------------------------------- launcher ----
extern "C" void kernel_launch(void* const* d_in, const int* in_sizes, int n_in,
                              void* d_out, int out_size, void* d_ws, size_t ws_size,
                              hipStream_t stream) {
    const float* x   = (const float*)d_in[0];
    const int*   ei  = (const int*)d_in[1];
    const float* W1  = (const float*)d_in[2];
    const float* b1  = (const float*)d_in[3];
    const float* W2  = (const float*)d_in[4];
    const float* b2  = (const float*)d_in[5];
    const float* Wfc = (const float*)d_in[6];
    const float* bfc = (const float*)d_in[7];
    float* out = (float*)d_out;

    const int IN_DIM = 256;
    const int HID = 64;
    const int n  = in_sizes[0] / IN_DIM;  // 50000
    const int ne = in_sizes[1] / 2;       // 800000
    const int* row = ei;
    const int* col = ei + ne;

    // Workspace layout: dinv | bufA | bufB
    char* ws = (char*)d_ws;
    float* dinv = (float*)ws;
    size_t off    = (((size_t)n * 4) + 255) & ~(size_t)255;
    size_t hbytes = (((size_t)n * HID * 4) + 255) & ~(size_t)255;
    float* bufA = (float*)(ws + off);
    float* bufB = (float*)(ws + off + hbytes);

    const dim3 blk(256);
    auto nb = [](long long t) { return dim3((unsigned)((t + 255) / 256)); };

    // Degrees + dinv
    k_fill_deg<<<nb(n), blk, 0, stream>>>(dinv, n);
    k_count_deg<<<nb(ne), blk, 0, stream>>>(col, dinv, ne);
    k_dinv<<<nb(n), blk, 0, stream>>>(dinv, n);

    const int tiles   = (n + 15) / 16;
    const int gblocks = (tiles + 7) / 8;  // 8 waves (16-row tiles) per block

    // Layer 1: gemm -> aggregate -> bias+relu
    k_gemm_wmma_n64<<<gblocks, 256, IN_DIM * 64 * sizeof(float), stream>>>(x, W1, bufA, n, IN_DIM);
    k_selfloop<<<nb((long long)n * 64), blk, 0, stream>>>(bufA, dinv, bufB, n);
    k_scatter<<<nb((long long)ne * 64), blk, 0, stream>>>(row, col, dinv, bufA, bufB, ne);
    k_bias_relu<<<nb((long long)n * 64), blk, 0, stream>>>(bufB, b1, n);

    // Layer 2
    k_gemm_wmma_n64<<<gblocks, 256, HID * 64 * sizeof(float), stream>>>(bufB, W2, bufA, n, HID);
    k_selfloop<<<nb((long long)n * 64), blk, 0, stream>>>(bufA, dinv, bufB, n);
    k_scatter<<<nb((long long)ne * 64), blk, 0, stream>>>(row, col, dinv, bufA, bufB, ne);
    k_bias_relu<<<nb((long long)n * 64), blk, 0, stream>>>(bufB, b2, n);

    // Final FC
    k_fc<<<nb(n), blk, 0, stream>>>(bufB, Wfc, bfc, out, n);
}